// FieldNet_45904610459864
// MI455X (gfx1250) — compile-verified
//
#include <hip/hip_runtime.h>
#include <math.h>

#define N_TRACK 50000
#define N_LANE  50000
#define N_ATT   (N_TRACK + N_LANE)
#define NPT     1000000
#define NPL     500000
#define E_ATT   1600000
#define HEADS   30
#define HPAD    64
#define EPSV    1e-5f
#define SLOPE   0.2f

typedef __attribute__((ext_vector_type(2))) float v2f;
typedef __attribute__((ext_vector_type(8))) float v8f;

__device__ __forceinline__ int lower_bound_i(const int* __restrict__ arr, int n, int val) {
    int lo = 0, hi = n;
    while (lo < hi) {
        int mid = (lo + hi) >> 1;
        if (arr[mid] < val) lo = mid + 1; else hi = mid;
    }
    return lo;
}

// ---------------- Track: MLP(4->4)+LN+ReLU -> segment max -> Linear(4->10) ----------------
__global__ void track_kernel(const float* __restrict__ pf, const int* __restrict__ pid,
                             const float* __restrict__ W, const float* __restrict__ b,
                             const float* __restrict__ g, const float* __restrict__ beta,
                             const float* __restrict__ oW, const float* __restrict__ ob,
                             float* __restrict__ feats) {
    int n = blockIdx.x * blockDim.x + threadIdx.x;
    if (n >= N_TRACK) return;
    int lo = lower_bound_i(pid, NPT, n);
    int hi = lower_bound_i(pid, NPT, n + 1);
    float w[16], bb[4], gg[4], be[4];
#pragma unroll
    for (int i = 0; i < 16; i++) w[i] = W[i];
#pragma unroll
    for (int i = 0; i < 4; i++) { bb[i] = b[i]; gg[i] = g[i]; be[i] = beta[i]; }
    float pool[4] = {0.f, 0.f, 0.f, 0.f};  // ReLU outputs >= 0; empty segment -> 0
    for (int p = lo; p < hi; p++) {
        float x0 = pf[p * 4 + 0], x1 = pf[p * 4 + 1], x2 = pf[p * 4 + 2], x3 = pf[p * 4 + 3];
        float h[4];
#pragma unroll
        for (int j = 0; j < 4; j++)
            h[j] = x0 * w[0 * 4 + j] + x1 * w[1 * 4 + j] + x2 * w[2 * 4 + j] + x3 * w[3 * 4 + j] + bb[j];
        float mu = 0.25f * (h[0] + h[1] + h[2] + h[3]);
        float var = 0.f;
#pragma unroll
        for (int j = 0; j < 4; j++) { float d = h[j] - mu; var += d * d; }
        var *= 0.25f;
        float inv = rsqrtf(var + EPSV);
#pragma unroll
        for (int j = 0; j < 4; j++) {
            float r = (h[j] - mu) * inv * gg[j] + be[j];
            r = r > 0.f ? r : 0.f;
            pool[j] = fmaxf(pool[j], r);
        }
    }
#pragma unroll
    for (int c = 0; c < 10; c++) {
        float s = ob[c];
#pragma unroll
        for (int j = 0; j < 4; j++) s += pool[j] * oW[j * 10 + c];
        feats[n * 10 + c] = s;
    }
}

// ---------------- Lane: MLP(2->2)+LN+ReLU -> segment max -> Linear(2->10) ----------------
__global__ void lane_kernel(const float* __restrict__ pf, const int* __restrict__ pid,
                            const float* __restrict__ W, const float* __restrict__ b,
                            const float* __restrict__ g, const float* __restrict__ beta,
                            const float* __restrict__ oW, const float* __restrict__ ob,
                            float* __restrict__ feats) {
    int n = blockIdx.x * blockDim.x + threadIdx.x;
    if (n >= N_LANE) return;
    int lo = lower_bound_i(pid, NPL, n);
    int hi = lower_bound_i(pid, NPL, n + 1);
    float w00 = W[0], w01 = W[1], w10 = W[2], w11 = W[3];
    float b0 = b[0], b1 = b[1], g0 = g[0], g1 = g[1], be0 = beta[0], be1 = beta[1];
    float p0 = 0.f, p1 = 0.f;
    for (int p = lo; p < hi; p++) {
        float x0 = pf[p * 2 + 0], x1 = pf[p * 2 + 1];
        float h0 = x0 * w00 + x1 * w10 + b0;
        float h1 = x0 * w01 + x1 * w11 + b1;
        float mu = 0.5f * (h0 + h1);
        float d0 = h0 - mu, d1 = h1 - mu;
        float var = 0.5f * (d0 * d0 + d1 * d1);
        float inv = rsqrtf(var + EPSV);
        float r0 = d0 * inv * g0 + be0; r0 = r0 > 0.f ? r0 : 0.f;
        float r1 = d1 * inv * g1 + be1; r1 = r1 > 0.f ? r1 : 0.f;
        p0 = fmaxf(p0, r0);
        p1 = fmaxf(p1, r1);
    }
#pragma unroll
    for (int c = 0; c < 10; c++)
        feats[(N_TRACK + n) * 10 + c] = p0 * oW[0 * 10 + c] + p1 * oW[1 * 10 + c] + ob[c];
}

// ---------------- h = feats @ fc_W via V_WMMA_F32_16X16X4_F32 ----------------
// One wave per 16-row tile. N=60 padded to 4 n-tiles of 16, K=10 padded to 3 k-steps of 4.
__global__ void fc_wmma_kernel(const float* __restrict__ feats, const float* __restrict__ Wfc,
                               float* __restrict__ hbuf) {
    int wave = (blockIdx.x * blockDim.x + threadIdx.x) >> 5;
    int lane = threadIdx.x & 31;
    int m0 = wave * 16;
    if (m0 >= N_ATT) return;              // wave-uniform: EXEC stays all-ones for WMMA
    int lm = lane & 15;
    int khalf = lane >> 4;                // A/B 16x4 layout: lanes 16-31 hold K+2,K+3
    int mrow = m0 + lm;
#pragma unroll
    for (int nt = 0; nt < 4; nt++) {
        int ncol = nt * 16 + lm;
        v8f c = {};
#pragma unroll
        for (int ks = 0; ks < 3; ks++) {
            int ka = ks * 4 + khalf * 2;
            v2f a, bm;
            a.x = (ka < 10)     ? feats[mrow * 10 + ka]     : 0.f;
            a.y = (ka + 1 < 10) ? feats[mrow * 10 + ka + 1] : 0.f;
            bm.x = (ka < 10 && ncol < 60)     ? Wfc[ka * 60 + ncol]       : 0.f;
            bm.y = (ka + 1 < 10 && ncol < 60) ? Wfc[(ka + 1) * 60 + ncol] : 0.f;
            c = __builtin_amdgcn_wmma_f32_16x16x4_f32(
                    /*neg_a=*/false, a, /*neg_b=*/false, bm,
                    /*c_mod=*/(short)0, c, /*reuse_a=*/false, /*reuse_b=*/false);
        }
#pragma unroll
        for (int r = 0; r < 8; r++)       // C/D layout: VGPR r -> rows r (lanes 0-15), r+8 (16-31)
            hbuf[(m0 + r + khalf * 8) * HPAD + nt * 16 + lm] = c[r];
    }
}

// ---------------- el/er: per (node, head) attention logit halves ----------------
__global__ void eler_kernel(const float* __restrict__ hbuf, const float* __restrict__ al,
                            const float* __restrict__ ar, float* __restrict__ el,
                            float* __restrict__ er) {
    int tid = blockIdx.x * blockDim.x + threadIdx.x;
    if (tid >= N_ATT * HEADS) return;
    int n = tid / HEADS, hd = tid % HEADS;
    float h0 = hbuf[n * HPAD + 2 * hd], h1 = hbuf[n * HPAD + 2 * hd + 1];
    el[tid] = h0 * al[2 * hd] + h1 * al[2 * hd + 1];
    er[tid] = h0 * ar[2 * hd] + h1 * ar[2 * hd + 1];
}

// ---------------- CSR offsets into sorted att_dst (only dst < N_TRACK matter) ----------------
__global__ void offs_kernel(const int* __restrict__ adst, int* __restrict__ offs) {
    int tid = blockIdx.x * blockDim.x + threadIdx.x;
    if (tid > N_TRACK) return;
    offs[tid] = lower_bound_i(adst, E_ATT, tid);
}

// ---------------- GAT edge softmax + aggregate, one thread per (dst, head) ----------------
__global__ void gat_kernel(const int* __restrict__ asrc, const float* __restrict__ el,
                           const float* __restrict__ er, const float* __restrict__ hbuf,
                           const int* __restrict__ offs, const float* __restrict__ bias,
                           float* __restrict__ out) {
    int tid = blockIdx.x * blockDim.x + threadIdx.x;
    if (tid >= N_TRACK * HEADS) return;
    int n = tid / HEADS, hd = tid % HEADS;
    int lo = offs[n], hi = offs[n + 1];
    float b0 = bias[2 * hd], b1 = bias[2 * hd + 1];
    if (lo >= hi) {  // no in-edges: segment_sum == 0
        out[n * 60 + 2 * hd] = b0;
        out[n * 60 + 2 * hd + 1] = b1;
        return;
    }
    float ern = er[n * HEADS + hd];
    float m = -INFINITY;
    for (int e = lo; e < hi; e++) {
        int s = asrc[e];
        float v = el[s * HEADS + hd] + ern;
        v = v > 0.f ? v : SLOPE * v;
        m = fmaxf(m, v);
    }
    float den = 0.f, a0 = 0.f, a1 = 0.f;
    for (int e = lo; e < hi; e++) {
        int s = asrc[e];
        float v = el[s * HEADS + hd] + ern;
        v = v > 0.f ? v : SLOPE * v;
        float ee = __expf(v - m);
        den += ee;
        a0 += ee * hbuf[s * HPAD + 2 * hd];
        a1 += ee * hbuf[s * HPAD + 2 * hd + 1];
    }
    float inv = 1.f / den;
    out[n * 60 + 2 * hd] = a0 * inv + b0;
    out[n * 60 + 2 * hd + 1] = a1 * inv + b1;
}

extern "C" void kernel_launch(void* const* d_in, const int* in_sizes, int n_in,
                              void* d_out, int out_size, void* d_ws, size_t ws_size,
                              hipStream_t stream) {
    const float* track_pf = (const float*)d_in[0];
    const float* lane_pf  = (const float*)d_in[1];
    const int*   pt_id    = (const int*)d_in[2];
    const int*   pl_id    = (const int*)d_in[3];
    const int*   att_src  = (const int*)d_in[4];
    const int*   att_dst  = (const int*)d_in[5];
    const float* t_W    = (const float*)d_in[6];
    const float* t_b    = (const float*)d_in[7];
    const float* t_g    = (const float*)d_in[8];
    const float* t_beta = (const float*)d_in[9];
    const float* t_oW   = (const float*)d_in[10];
    const float* t_ob   = (const float*)d_in[11];
    const float* l_W    = (const float*)d_in[12];
    const float* l_b    = (const float*)d_in[13];
    const float* l_g    = (const float*)d_in[14];
    const float* l_beta = (const float*)d_in[15];
    const float* l_oW   = (const float*)d_in[16];
    const float* l_ob   = (const float*)d_in[17];
    const float* fc_W   = (const float*)d_in[18];
    const float* attn_l = (const float*)d_in[19];
    const float* attn_r = (const float*)d_in[20];
    const float* g_bias = (const float*)d_in[21];
    float* out = (float*)d_out;

    // Workspace layout (~54 MB)
    float* feats = (float*)d_ws;                       // N_ATT*10
    float* hbuf  = feats + (size_t)N_ATT * 10;         // N_ATT*HPAD
    float* el    = hbuf + (size_t)N_ATT * HPAD;        // N_ATT*HEADS
    float* er    = el + (size_t)N_ATT * HEADS;         // N_ATT*HEADS
    int*   offs  = (int*)(er + (size_t)N_ATT * HEADS); // N_TRACK+1

    const int B = 256;
    track_kernel<<<(N_TRACK + B - 1) / B, B, 0, stream>>>(
        track_pf, pt_id, t_W, t_b, t_g, t_beta, t_oW, t_ob, feats);
    lane_kernel<<<(N_LANE + B - 1) / B, B, 0, stream>>>(
        lane_pf, pl_id, l_W, l_b, l_g, l_beta, l_oW, l_ob, feats);

    int waves = N_ATT / 16;                 // 6250 tiles, 8 waves per 256-thread block
    int fc_blocks = (waves + 7) / 8;
    fc_wmma_kernel<<<fc_blocks, B, 0, stream>>>(feats, fc_W, hbuf);

    eler_kernel<<<((size_t)N_ATT * HEADS + B - 1) / B, B, 0, stream>>>(hbuf, attn_l, attn_r, el, er);
    offs_kernel<<<(N_TRACK + 1 + B - 1) / B, B, 0, stream>>>(att_dst, offs);
    gat_kernel<<<((size_t)N_TRACK * HEADS + B - 1) / B, B, 0, stream>>>(
        att_src, el, er, hbuf, offs, g_bias, out);
}